// NoiseAwareRelationInference_5411658793123
// MI455X (gfx1250) — compile-verified
//
#include <hip/hip_runtime.h>
#include <math.h>

#define N 8192
#define Fdim 128
#define Ddim 64
#define SLOPE 0.01f

typedef __attribute__((ext_vector_type(2))) float v2f;
typedef __attribute__((ext_vector_type(8))) float v8f;

// ---- workspace layout (float offsets) ----
// [0,256)        wvec = [w1 (128) | w2 (128)]
// [256,8448)     s1[N]
// [8448,16640)   s2[N]
// [16640,24832)  partial sums  (per row-block)
// [24832,33024)  partial sumsq (per row-block)
// [33024]        mean
// [33025]        inv_sig
#define WS_W    0
#define WS_S1   256
#define WS_S2   (256 + N)
#define WS_PS   (256 + 2*N)
#define WS_PQ   (256 + 3*N)
#define WS_STAT (256 + 4*N)

// Kernel 1: wvec[f] = sum_d W_r[f,d] * a_r[d]   (f in [0,256))
__global__ __launch_bounds__(256) void prep_kernel(const float* __restrict__ W_r,
                                                   const float* __restrict__ a_r,
                                                   float* __restrict__ ws) {
    int f = threadIdx.x;
    float acc = 0.f;
#pragma unroll 8
    for (int d = 0; d < Ddim; ++d) acc += W_r[f * Ddim + d] * a_r[d];
    ws[WS_W + f] = acc;
}

// Kernel 2: s1 = X @ w1, s2 = X @ w2 via V_WMMA_F32_16X16X4_F32.
// Each wave computes 16 rows. B fragment: col0 = w1, col1 = w2, rest 0.
__global__ __launch_bounds__(256) void matvec_wmma_kernel(const float* __restrict__ X,
                                                          const float* __restrict__ ws) {
    const float* __restrict__ wvec = ws + WS_W;
    float* __restrict__ s1 = (float*)(ws + WS_S1);
    float* __restrict__ s2 = (float*)(ws + WS_S2);

    const int lane   = threadIdx.x & 31;
    const int waveG  = blockIdx.x * (blockDim.x >> 5) + (threadIdx.x >> 5);
    const int rowB   = waveG * 16;           // 16 rows per wave
    const int nlane  = lane & 15;            // M (for A) / N (for B)
    const int khalf  = (lane >> 4) * 2;      // lanes 0-15: K 0,1 ; lanes 16-31: K 2,3
    const int arow   = rowB + nlane;

    // branch-free B column select: cols >=2 are zero
    const int   bcol = (nlane < 2) ? nlane : 0;
    const float bmsk = (nlane < 2) ? 1.f : 0.f;
    const float* bp  = wvec + bcol * Fdim;

    v8f c = {};
    for (int k = 0; k < Fdim; k += 4) {
        v2f a = *(const v2f*)(X + (size_t)arow * Fdim + k + khalf);
        v2f b;
        b.x = bp[k + khalf]     * bmsk;
        b.y = bp[k + khalf + 1] * bmsk;
        c = __builtin_amdgcn_wmma_f32_16x16x4_f32(false, a, false, b,
                                                  (short)0, c, false, false);
    }
    // D layout: VGPR r, lanes 0-15 -> (M=r, N=lane); lanes 16-31 -> (M=8+r, N=lane-16)
    if (nlane < 2) {
        float* dst = (nlane == 0) ? s1 : s2;
        int m = rowB + ((lane >> 4) ? 8 : 0);
#pragma unroll
        for (int r = 0; r < 8; ++r) dst[m + r] = c[r];
    }
}

// Kernel 3: per-row partial Σ and Σx² of leaky_relu(s1_i + s2_j)
__global__ __launch_bounds__(256) void stats_kernel(float* __restrict__ ws) {
    __shared__ float s2l[N];
    __shared__ float redA[256];
    __shared__ float redB[256];
    const float* s1 = ws + WS_S1;
    const float* s2 = ws + WS_S2;
    const int tid = threadIdx.x;
    const int i   = blockIdx.x;

    for (int j = tid; j < N; j += 256) s2l[j] = s2[j];
    __syncthreads();

    const float si = s1[i];
    float sum = 0.f, sq = 0.f;
    for (int j = tid; j < N; j += 256) {
        float x = si + s2l[j];
        float r = (x > 0.f) ? x : SLOPE * x;
        sum += r;
        sq  += r * r;
    }
    redA[tid] = sum; redB[tid] = sq;
    __syncthreads();
    for (int s = 128; s > 0; s >>= 1) {
        if (tid < s) { redA[tid] += redA[tid + s]; redB[tid] += redB[tid + s]; }
        __syncthreads();
    }
    if (tid == 0) { ws[WS_PS + i] = redA[0]; ws[WS_PQ + i] = redB[0]; }
}

// Kernel 4: deterministic fold of partials in double -> mean, 1/(std+1e-5)
__global__ __launch_bounds__(256) void finalize_kernel(float* __restrict__ ws) {
    __shared__ double rA[256];
    __shared__ double rB[256];
    const int tid = threadIdx.x;
    double s = 0.0, q = 0.0;
    for (int b = tid; b < N; b += 256) { s += (double)ws[WS_PS + b]; q += (double)ws[WS_PQ + b]; }
    rA[tid] = s; rB[tid] = q;
    __syncthreads();
    for (int k = 128; k > 0; k >>= 1) {
        if (tid < k) { rA[tid] += rA[tid + k]; rB[tid] += rB[tid + k]; }
        __syncthreads();
    }
    if (tid == 0) {
        const double M = (double)N * (double)N;
        double sum = rA[0], sumsq = rB[0];
        double mean = sum / M;
        double var  = (sumsq - sum * sum / M) / (M - 1.0);  // ddof=1
        float  sig  = (float)sqrt(var);
        ws[WS_STAT + 0] = (float)mean;
        ws[WS_STAT + 1] = 1.0f / (sig + 1e-5f);
    }
}

// Kernel 5: row softmax + threshold (2/N), streamed 0/1 output.
__global__ __launch_bounds__(256) void softmax_thresh_kernel(const float* __restrict__ ws,
                                                             float* __restrict__ out) {
    __shared__ float tl[N];
    __shared__ float red[256];
    const float* s1 = ws + WS_S1;
    const float* s2 = ws + WS_S2;
    const int tid = threadIdx.x;
    const int i   = blockIdx.x;

    const float mean = ws[WS_STAT + 0];
    const float inv  = ws[WS_STAT + 1];
    const float si   = s1[i];

    float lmax = -3.4e38f;
    for (int j = tid; j < N; j += 256) {
        float x = si + s2[j];
        float r = (x > 0.f) ? x : SLOPE * x;
        float t = (r - mean) * inv;
        tl[j] = t;
        lmax = fmaxf(lmax, t);
    }
    red[tid] = lmax;
    __syncthreads();
    for (int s = 128; s > 0; s >>= 1) {
        if (tid < s) red[tid] = fmaxf(red[tid], red[tid + s]);
        __syncthreads();
    }
    const float mx = red[0];
    __syncthreads();

    float lsum = 0.f;
    for (int j = tid; j < N; j += 256) lsum += __expf(tl[j] - mx);
    red[tid] = lsum;
    __syncthreads();
    for (int s = 128; s > 0; s >>= 1) {
        if (tid < s) red[tid] += red[tid + s];
        __syncthreads();
    }
    const float Z = red[0];

    // E_ij = exp(t - mx)/Z >= 2/N  <=>  t >= mx + log((2/N)*Z)
    const float thr = 2.0f / (float)N;
    const float logcut = mx + __logf(thr * Z);

    float* __restrict__ orow = out + (size_t)i * N;
    for (int j = tid; j < N; j += 256) {
        float v = (tl[j] >= logcut) ? 1.0f : 0.0f;
        __builtin_nontemporal_store(v, orow + j);
    }
}

extern "C" void kernel_launch(void* const* d_in, const int* in_sizes, int n_in,
                              void* d_out, int out_size, void* d_ws, size_t ws_size,
                              hipStream_t stream) {
    const float* X   = (const float*)d_in[0];   // [8192,128]
    const float* W_r = (const float*)d_in[1];   // [256,64]
    const float* a_r = (const float*)d_in[2];   // [64]
    float* out = (float*)d_out;                 // [8192,8192]
    float* ws  = (float*)d_ws;

    prep_kernel<<<1, 256, 0, stream>>>(W_r, a_r, ws);
    // 8192 rows / 16 rows-per-wave = 512 waves; 8 waves per 256-thread block -> 64 blocks
    matvec_wmma_kernel<<<64, 256, 0, stream>>>(X, ws);
    stats_kernel<<<N, 256, 0, stream>>>(ws);
    finalize_kernel<<<1, 256, 0, stream>>>(ws);
    softmax_thresh_kernel<<<N, 256, 0, stream>>>(ws, out);
}